// PETP_Quadratic_790273982668
// MI455X (gfx1250) — compile-verified
//
#include <hip/hip_runtime.h>
#include <hip/hip_bf16.h>

typedef float v2f __attribute__((ext_vector_type(2)));
typedef float v4f __attribute__((ext_vector_type(4)));
typedef float v8f __attribute__((ext_vector_type(8)));

#define LDS_ROW 129              // 128 channels + 1 pad dword (conflict-free row stride)
#define LDS_VAR (48 * LDS_ROW)   // per-variant block: 48 rows

// ---------------------------------------------------------------------------
// Weight repack: source W[t][tau][u][v][w] (t = variant array) -> packed per-lane
// WMMA-B fragments in d_ws, with norm (and 1/sqrt3 for tau==1) pre-folded.
// Packed float4 index: (((t*4 + tau)*32 + u)*8 + vb)*32 + lane
//   lane: kk = (lane>>4)*2, nl = lane&15, v0 = vb*4
//   elems: {W[v0+kk][nl], W[v0+kk+1][nl], W[v0+kk][16+nl], W[v0+kk+1][16+nl]}
// ---------------------------------------------------------------------------
__global__ __launch_bounds__(256) void petp_repack_kernel(
    const float* __restrict__ W_self, const float* __restrict__ W_flavor,
    const float* __restrict__ W_nunubar, const float* __restrict__ W_all,
    float* __restrict__ wp)
{
    const int p4 = blockIdx.x * 256 + threadIdx.x;   // 0 .. 131071
    const int lane = p4 & 31;
    const int vb   = (p4 >> 5) & 7;
    const int u    = (p4 >> 8) & 31;
    const int tau  = (p4 >> 13) & 3;
    const int t    = p4 >> 15;

    const float* Wt[4] = {W_self, W_flavor, W_nunubar, W_all};
    const int kk = (lane >> 4) * 2;
    const int nl = lane & 15;
    const int v0 = vb * 4 + kk;

    const float norm = 0.022097086912079608f;        // 1/(32*sqrt(2))
    const float inv3 = 0.57735026918962576f;         // 1/sqrt(3) for the W1 einsum
    const float scale = (tau == 1) ? norm * inv3 : norm;

    const float* src = Wt[t] + ((size_t)(tau * 32 + u)) * 1024;  // [v][w] block
    v4f o;
    o.x = scale * src[(v0    ) * 32 + nl];
    o.y = scale * src[(v0 + 1) * 32 + nl];
    o.z = scale * src[(v0    ) * 32 + 16 + nl];
    o.w = scale * src[(v0 + 1) * 32 + 16 + nl];
    ((v4f*)wp)[p4] = o;
}

// ---------------------------------------------------------------------------
// Main kernel: one workgroup = 8 batches = 48 rows (3 waves x 16-row M-tiles).
// y[n, 0:128], n = b*6 + i*3 + j.  GEMM with on-the-fly quadratic features.
// ---------------------------------------------------------------------------
__global__ __launch_bounds__(96) void petp_quadratic_kernel(
    const float* __restrict__ x,
    const float* __restrict__ wp,     // packed weights (norm pre-folded)
    float* __restrict__ out)
{
    extern __shared__ float lds[];    // 4 variants * 48 rows * 129 floats = 99072 B

    const int tid = threadIdx.x;
    const int b0  = blockIdx.x * 8;

    // ---------------- Phase 1: pe_inputs -> LDS ----------------
    for (int idx = tid; idx < 8 * 128; idx += 96) {
        const int bl = idx >> 7;
        const int c  = idx & 127;
        const float* xb = x + (size_t)(b0 + bl) * 768 + c;  // (2,3,128) per batch
        const float x00 = xb[0],   x01 = xb[128], x02 = xb[256];
        const float x10 = xb[384], x11 = xb[512], x12 = xb[640];
        const float rs0 = x00 + x01 + x02, rs1 = x10 + x11 + x12;
        const float cs0 = x00 + x10, cs1 = x01 + x11, cs2 = x02 + x12;
        const float ts  = rs0 + rs1;
        const float xx[2][3] = {{x00, x01, x02}, {x10, x11, x12}};
        const float rs[2] = {rs0, rs1};
        const float cs[3] = {cs0, cs1, cs2};
        #pragma unroll
        for (int i = 0; i < 2; ++i) {
            #pragma unroll
            for (int j = 0; j < 3; ++j) {
                const float s  = xx[i][j];
                const float xf = 0.5f * (rs[i] - s);              // /(3-1)
                const float xn = cs[j] - s;                       // /(2-1)
                const float xa = 0.5f * (ts - rs[i] - cs[j] + s); // /2
                const int row = bl * 6 + i * 3 + j;
                lds[0 * LDS_VAR + row * LDS_ROW + c] = s;
                lds[1 * LDS_VAR + row * LDS_ROW + c] = xf;
                lds[2 * LDS_VAR + row * LDS_ROW + c] = xn;
                lds[3 * LDS_VAR + row * LDS_ROW + c] = xa;
            }
        }
    }
    __syncthreads();

    // ---------------- Phase 2: WMMA GEMM ----------------
    const int wave = tid >> 5;
    const int lane = tid & 31;
    const int nl   = lane & 15;   // A row within tile / D column
    const int hi   = lane >> 4;
    const int kk   = hi * 2;      // A VGPRs hold K = kk, kk+1

    const v8f vzero = {0.f, 0.f, 0.f, 0.f, 0.f, 0.f, 0.f, 0.f};
    v8f acc[8];                   // [0,1]=y_s halves; [2+2k,3+2k]=y_v[:,k] halves
    #pragma unroll
    for (int a = 0; a < 8; ++a) acc[a] = vzero;

    #pragma unroll 1
    for (int t = 0; t < 4; ++t) {
        const float* rowp = lds + t * LDS_VAR + (wave * 16 + nl) * LDS_ROW;
        // per-variant packed base, lane offset folded in (float4 units)
        const v4f* wpt = (const v4f*)wp + (size_t)t * 32768 + lane;
        #pragma unroll 1
        for (int vb = 0; vb < 8; ++vb) {
            const int va = vb * 4 + kk;
            // v-side features: hoisted out of the u loop
            const float sA = rowp[va],          sB = rowp[va + 1];
            const float a0 = rowp[32 + va * 3 + 0];
            const float a1 = rowp[32 + va * 3 + 1];
            const float a2 = rowp[32 + va * 3 + 2];
            const float b0 = rowp[32 + va * 3 + 3];
            const float b1 = rowp[32 + va * 3 + 4];
            const float b2 = rowp[32 + va * 3 + 5];
            const v4f* wpv = wpt + vb * 32;
            #pragma unroll 1
            for (int u = 0; u < 32; ++u) {
                const float su  = rowp[u];
                const float vu0 = rowp[32 + u * 3 + 0];
                const float vu1 = rowp[32 + u * 3 + 1];
                const float vu2 = rowp[32 + u * 3 + 2];

                // A fragments (16x4 f32: lane&15 = row, 2 VGPRs = K kk,kk+1)
                const v2f A0 = {su * sA, su * sB};
                const v2f A1 = {vu0 * a0 + vu1 * a1 + vu2 * a2,    // 1/sqrt3 folded in W1
                                vu0 * b0 + vu1 * b1 + vu2 * b2};
                const v2f A2[3] = {{su * a0, su * b0},
                                   {su * a1, su * b1},
                                   {su * a2, su * b2}};
                const v2f A3[3] = {{vu0 * sA, vu0 * sB},
                                   {vu1 * sA, vu1 * sB},
                                   {vu2 * sA, vu2 * sB}};

                // B fragments: one b128 per tau (tau stride = 8192 float4 = const offset)
                const v4f* wpu = wpv + u * 256;
                const v4f w0 = wpu[0];
                const v4f w1 = wpu[8192];
                const v4f w2 = wpu[16384];
                const v4f w3 = wpu[24576];
                const v2f B00 = {w0.x, w0.y}, B01 = {w0.z, w0.w};
                const v2f B10 = {w1.x, w1.y}, B11 = {w1.z, w1.w};
                const v2f B20 = {w2.x, w2.y}, B21 = {w2.z, w2.w};
                const v2f B30 = {w3.x, w3.y}, B31 = {w3.z, w3.w};

                acc[0] = __builtin_amdgcn_wmma_f32_16x16x4_f32(false, A0, false, B00, (short)0, acc[0], false, false);
                acc[1] = __builtin_amdgcn_wmma_f32_16x16x4_f32(false, A0, false, B01, (short)0, acc[1], false, false);
                acc[0] = __builtin_amdgcn_wmma_f32_16x16x4_f32(false, A1, false, B10, (short)0, acc[0], false, false);
                acc[1] = __builtin_amdgcn_wmma_f32_16x16x4_f32(false, A1, false, B11, (short)0, acc[1], false, false);
                #pragma unroll
                for (int k = 0; k < 3; ++k) {
                    acc[2 + 2 * k] = __builtin_amdgcn_wmma_f32_16x16x4_f32(false, A2[k], false, B20, (short)0, acc[2 + 2 * k], false, false);
                    acc[3 + 2 * k] = __builtin_amdgcn_wmma_f32_16x16x4_f32(false, A2[k], false, B21, (short)0, acc[3 + 2 * k], false, false);
                    acc[2 + 2 * k] = __builtin_amdgcn_wmma_f32_16x16x4_f32(false, A3[k], false, B30, (short)0, acc[2 + 2 * k], false, false);
                    acc[3 + 2 * k] = __builtin_amdgcn_wmma_f32_16x16x4_f32(false, A3[k], false, B31, (short)0, acc[3 + 2 * k], false, false);
                }
            }
        }
    }

    // ---------------- Store (norm pre-folded; C/D layout: M = r + 8*hi, N = nl) ----------------
    const int n0 = blockIdx.x * 48 + wave * 16;
    #pragma unroll
    for (int r = 0; r < 8; ++r) {
        const int m = r + hi * 8;
        float* orow = out + (size_t)(n0 + m) * 128;
        orow[nl]      = acc[0][r];
        orow[16 + nl] = acc[1][r];
        #pragma unroll
        for (int k = 0; k < 3; ++k) {
            orow[32 + nl * 3 + k]        = acc[2 + 2 * k][r];
            orow[32 + (16 + nl) * 3 + k] = acc[3 + 2 * k][r];
        }
    }
}

extern "C" void kernel_launch(void* const* d_in, const int* in_sizes, int n_in,
                              void* d_out, int out_size, void* d_ws, size_t ws_size,
                              hipStream_t stream) {
    const float* x  = (const float*)d_in[0];
    const float* Ws = (const float*)d_in[1];
    const float* Wf = (const float*)d_in[2];
    const float* Wn = (const float*)d_in[3];
    const float* Wa = (const float*)d_in[4];
    float* out = (float*)d_out;
    float* wp  = (float*)d_ws;                   // 2 MB packed weights

    // 1) repack + pre-scale weights into WMMA-B lane layout (131072 float4)
    petp_repack_kernel<<<512, 256, 0, stream>>>(Ws, Wf, Wn, Wa, wp);

    // 2) main fused kernel
    const int B = in_sizes[0] / (2 * 3 * 128);   // 16384
    const int blocks = B / 8;                    // 2048
    const size_t lds_bytes = 4u * 48u * LDS_ROW * sizeof(float);  // 99072 B
    petp_quadratic_kernel<<<blocks, 96, lds_bytes, stream>>>(x, wp, out);
}